// ExpertConv2d_26766236189235
// MI455X (gfx1250) — compile-verified
//
#include <hip/hip_runtime.h>

typedef float v2f __attribute__((ext_vector_type(2)));
typedef float v8f __attribute__((ext_vector_type(8)));
typedef unsigned int u32x4 __attribute__((ext_vector_type(4)));
typedef int i32x4 __attribute__((ext_vector_type(4)));
typedef int i32x8 __attribute__((ext_vector_type(8)));

#define NPATCH 512
#define CIN 16
#define COUT 16
#define PSZ 64
#define KEXP 5
#define KDIM 144           // CIN*3*3
#define KSTEPS 36          // KDIM/4
#define ROWS_PER_BLK 8
#define LROWS 10           // ROWS_PER_BLK + 2 halo rows
#define WPAD 68            // 1 + 64 + 3 zero-pad cols (stride 68)
#define AW_PER_PATCH 2304  // COUT * KDIM

// Compile-time tap base for flat K index k = (i, kh, kw):
// offset of x_lds element (i*LROWS + kh)*WPAD + kw; add rb = r*WPAD + nlane
// at runtime to get the per-lane B-operand address.
__host__ __device__ constexpr int tapbase(int k) {
  return ((k / 9) * LROWS + (k % 9) / 3) * WPAD + (k % 9) % 3;
}

// ---------------------------------------------------------------------------
// TDM issue helper: arity differs between toolchains (probe-confirmed).
// ---------------------------------------------------------------------------
__device__ __forceinline__ void tdm_load(u32x4 g0, i32x8 g1, i32x4 g2, i32x4 g3) {
#if defined(__clang_major__) && __clang_major__ >= 23
  i32x8 gz = {0, 0, 0, 0, 0, 0, 0, 0};
  __builtin_amdgcn_tensor_load_to_lds(g0, g1, g2, g3, gz, 0);
#else
  __builtin_amdgcn_tensor_load_to_lds(g0, g1, g2, g3, 0);
#endif
}

// ---------------------------------------------------------------------------
// Kernel 1: mix expert kernels per patch and pre-swizzle into WMMA A layout.
// A-operand layout (f32 16x16x4, ISA 7.12.2): element e = (s*2+vg)*32 + lane,
//   m = lane&15, ksub = 2*(lane>>4) + vg, k = 4*s + ksub.
// ---------------------------------------------------------------------------
__global__ void expertconv_mix_kernel(const float* __restrict__ v,
                                      const float* __restrict__ weight,
                                      const float* __restrict__ bias,
                                      float* __restrict__ ws_a,
                                      float* __restrict__ ws_b) {
  int n = blockIdx.x;                 // patch id
  int b = n >> 6, hh = (n >> 3) & 7, ww = n & 7;

  float vv[KEXP];
#pragma unroll
  for (int k = 0; k < KEXP; ++k)
    vv[k] = v[((b * KEXP + k) * 8 + hh) * 8 + ww];

  for (int e = threadIdx.x; e < AW_PER_PATCH; e += blockDim.x) {
    int lane = e & 31;
    int vg   = (e >> 5) & 1;
    int s    = e >> 6;
    int ksub = 2 * (lane >> 4) + vg;
    int k    = 4 * s + ksub;          // 0..143
    int m    = lane & 15;             // out channel
    int i    = k / 9;
    int r9   = k % 9;
    int kh   = r9 / 3, kw = r9 % 3;
    float sum = 0.f;
#pragma unroll
    for (int kk = 0; kk < KEXP; ++kk)
      sum += vv[kk] * weight[(((kk * COUT + m) * CIN + i) * 3 + kh) * 3 + kw];
    ws_a[(size_t)n * AW_PER_PATCH + e] = sum;
  }

  if (threadIdx.x < COUT) {
    float sum = 0.f;
#pragma unroll
    for (int kk = 0; kk < KEXP; ++kk)
      sum += vv[kk] * bias[kk * COUT + threadIdx.x];
    ws_b[n * COUT + threadIdx.x] = sum;
  }
}

// ---------------------------------------------------------------------------
// Kernel 2: per-patch conv as implicit GEMM on V_WMMA_F32_16X16X4_F32.
// Block = (patch, 8-row group), 256 threads = 8 waves; wave -> 1 output row,
// 4 col tiles of 16 pixels. Input slab + A matrix staged via TDM
// (tensor_load_to_lds) with hardware row padding into the 68-stride layout.
// K-loop fully unrolled: tap offsets are compile-time constants selected by
// lane half, so the straight-line body is ds_load + v_cndmask + v_wmma.
// ---------------------------------------------------------------------------
__global__ __launch_bounds__(256) void expertconv_wmma_kernel(
    const float* __restrict__ x,
    const float* __restrict__ ws_a,
    const float* __restrict__ ws_b,
    float* __restrict__ out) {
  __shared__ float x_lds[CIN * LROWS * WPAD];  // 43520 B
  __shared__ float a_lds[AW_PER_PATCH];        //  9216 B

  int n   = blockIdx.x >> 3;          // patch id
  int grp = blockIdx.x & 7;           // row group within patch
  int b = n >> 6, hh = (n >> 3) & 7, ww = n & 7;
  int gr0 = grp * ROWS_PER_BLK;
  int tid = threadIdx.x;

  // --- Pre-zero the regions TDM's padding skips: pad cols 0,65,66,67 ------
  for (int idx = tid; idx < CIN * LROWS * 4; idx += 256) {
    int ch  = idx / (LROWS * 4);
    int rem = idx % (LROWS * 4);
    int lr  = rem >> 2;
    int c   = rem & 3;
    int cp  = (c == 0) ? 0 : 64 + c;  // cols 0,65,66,67
    x_lds[(ch * LROWS + lr) * WPAD + cp] = 0.f;
  }
  // --- Pre-zero out-of-patch halo rows (patch-local zero padding) ---------
  if (grp == 0) {
    for (int idx = tid; idx < CIN * WPAD; idx += 256) {
      int ch = idx / WPAD, cp = idx % WPAD;
      x_lds[(ch * LROWS + 0) * WPAD + cp] = 0.f;
    }
  } else if (grp == 7) {
    for (int idx = tid; idx < CIN * WPAD; idx += 256) {
      int ch = idx / WPAD, cp = idx % WPAD;
      x_lds[(ch * LROWS + (LROWS - 1)) * WPAD + cp] = 0.f;
    }
  }

  // --- Wave 0 issues two TDM descriptors, waits, then block barrier -------
  if (tid < 32) {
    int r_start = (gr0 - 1 < 0) ? 0 : gr0 - 1;
    int r_end   = (gr0 + ROWS_PER_BLK + 1 > PSZ) ? PSZ : (gr0 + ROWS_PER_BLK + 1);
    int nrows   = r_end - r_start;          // 9 or 10
    int row_off = r_start - (gr0 - 1);      // 0 or 1

    // x slab: 2D tile 64 cols x nrows rows, iterated over 16 channels.
    unsigned lds_x = (unsigned)(size_t)(void*)x_lds
                   + (unsigned)(row_off * WPAD * 4 + 4);  // skip pad col 0
    unsigned long long ga_x = (unsigned long long)(size_t)(const void*)
        (x + (((size_t)b * CIN) * 512 + (size_t)(hh * 64 + r_start)) * 512
           + (size_t)(ww * 64));
    u32x4 g0x = { 1u, lds_x, (unsigned)ga_x,
                  (unsigned)(ga_x >> 32) | 0x80000000u };   // type=2 (image)
    // d0: data_size=4B(2<<16) | iterate(1<<19) | pad(1<<20)
    //     | pad_interval=64dw(5<<22) | pad_amount=4dw(3<<25)
    i32x8 g1x = { (int)0x075A0000,
                  (int)(64u << 16),       // tensor_dim0 = 64
                  (int)(64u << 16),       // tensor_dim1 = 64
                  (int)(64u << 16),       // tile_dim0 = 64
                  nrows,                  // tile_dim1 = nrows, tile_dim2 = 0
                  512,                    // tensor_dim0_stride (elements)
                  0, 0 };
    i32x4 g2x = { 0,                      // tensor_dim2 (unused)
                  LROWS * WPAD,           // lds_addr_increment / iter (elems)
                  512 * 512,              // global_addr_increment / iter
                  (int)((unsigned)(CIN - 1) << 16) };  // iterate_count = 15
    i32x4 g3z = { 0, 0, 0, 0 };
    tdm_load(g0x, g1x, g2x, g3z);

    // A matrix: contiguous 2304-element 1D copy.
    unsigned lds_a = (unsigned)(size_t)(void*)a_lds;
    unsigned long long ga_a = (unsigned long long)(size_t)(const void*)
        (ws_a + (size_t)n * AW_PER_PATCH);
    u32x4 g0a = { 1u, lds_a, (unsigned)ga_a,
                  (unsigned)(ga_a >> 32) | 0x80000000u };
    i32x8 g1a = { (int)0x00020000,                          // data_size=4B
                  (int)((AW_PER_PATCH & 0xffff) << 16),     // tensor_dim0
                  (int)(1u << 16),                          // tensor_dim1 = 1
                  (int)((AW_PER_PATCH & 0xffff) << 16),     // tile_dim0
                  1,                                        // tile_dim1 = 1
                  AW_PER_PATCH,                             // dim0 stride
                  0, 0 };
    tdm_load(g0a, g1a, g3z, g3z);

    __builtin_amdgcn_s_wait_tensorcnt(0);
  }
  __syncthreads();

  int lane  = tid & 31;
  int r     = tid >> 5;               // wave id -> output row in group (0..7)
  int nlane = lane & 15;              // pixel within tile (N)
  int khalf = lane >> 4;
  int rb    = r * WPAD + nlane;       // loop-invariant per-lane base

  v8f acc[4];
#pragma unroll
  for (int t = 0; t < 4; ++t) acc[t] = (v8f){0.f,0.f,0.f,0.f,0.f,0.f,0.f,0.f};

#pragma unroll
  for (int s = 0; s < KSTEPS; ++s) {
    v2f A;
    A.x = a_lds[(s * 2 + 0) * 32 + lane];
    A.y = a_lds[(s * 2 + 1) * 32 + lane];

    // Compile-time tap bases for both lane halves (k = 4s + 2*khalf + vg).
    const int bA0 = tapbase(4 * s + 0);   // khalf=0, vg=0
    const int bB0 = tapbase(4 * s + 2);   // khalf=1, vg=0
    const int bA1 = tapbase(4 * s + 1);   // khalf=0, vg=1
    const int bB1 = tapbase(4 * s + 3);   // khalf=1, vg=1
    int bas0 = rb + (khalf ? bB0 : bA0);
    int bas1 = rb + (khalf ? bB1 : bA1);

#pragma unroll
    for (int t = 0; t < 4; ++t) {
      v2f B;
      B.x = x_lds[bas0 + 16 * t];
      B.y = x_lds[bas1 + 16 * t];
      acc[t] = __builtin_amdgcn_wmma_f32_16x16x4_f32(
          /*neg_a=*/false, A, /*neg_b=*/false, B,
          /*c_mod=*/(short)0, acc[t], /*reuse_a=*/false, /*reuse_b=*/false);
    }
  }

  // Mixed bias per output channel for this lane half.
  float bv[8];
#pragma unroll
  for (int j = 0; j < 8; ++j)
    bv[j] = ws_b[n * COUT + j + 8 * khalf];

  // C/D layout: VGPR j, lanes 0-15 -> M=j, lanes 16-31 -> M=j+8; N = lane&15.
#pragma unroll
  for (int t = 0; t < 4; ++t) {
#pragma unroll
    for (int j = 0; j < 8; ++j) {
      int o = j + 8 * khalf;
      size_t oidx = (((size_t)b * COUT + o) * 512 + (size_t)(hh * 64 + gr0 + r)) * 512
                    + (size_t)(ww * 64 + 16 * t + nlane);
      out[oidx] = acc[t][j] + bv[j];
    }
  }
}

extern "C" void kernel_launch(void* const* d_in, const int* in_sizes, int n_in,
                              void* d_out, int out_size, void* d_ws, size_t ws_size,
                              hipStream_t stream) {
  const float* x      = (const float*)d_in[0];
  const float* v      = (const float*)d_in[1];
  const float* weight = (const float*)d_in[2];
  const float* bias   = (const float*)d_in[3];
  float* out  = (float*)d_out;
  float* ws_a = (float*)d_ws;                          // 512*2304 floats
  float* ws_b = ws_a + (size_t)NPATCH * AW_PER_PATCH;  // 512*16 floats

  expertconv_mix_kernel<<<NPATCH, 64, 0, stream>>>(v, weight, bias, ws_a, ws_b);
  expertconv_wmma_kernel<<<NPATCH * 8, 256, 0, stream>>>(x, ws_a, ws_b, out);
}